// ClusterGCN_15556371546774
// MI455X (gfx1250) — compile-verified
//
#include <hip/hip_runtime.h>
#include <cstdint>
#include <cstddef>

// ---------------------------------------------------------------------------
// ClusterGCN (3-layer) for MI455X / gfx1250.
//  - gather/scatter via f32 global atomics (memory-bound part)
//  - dense layers via V_WMMA_F32_16X16X4_F32, branchless inner loop,
//    64x16 output block per wave (4 row-tiles share each weight fragment),
//    load-phase / wmma-phase structure so loads stay in flight across WMMAs
// ---------------------------------------------------------------------------

typedef __attribute__((ext_vector_type(2))) float v2f;
typedef __attribute__((ext_vector_type(8))) float v8f;

#define IN_CH  256
#define HID    256
#define OUT_CH 121
#define OUT_PAD 128   // OUT_CH padded to a multiple of 16

// deg[i] = 1.0 (self loop contribution)
__global__ void k_fill1(float* __restrict__ p, int n) {
    int i = blockIdx.x * blockDim.x + threadIdx.x;
    if (i < n) p[i] = 1.0f;
}

// deg[col[e]] += 1 for every edge
__global__ void k_deg_accum(const int* __restrict__ col, float* __restrict__ deg, int E) {
    int e = blockIdx.x * blockDim.x + threadIdx.x;
    if (e < E) atomicAdd(&deg[col[e]], 1.0f);
}

// deg_inv = 1/deg  (deg >= 1 always, so clip(.,1) is implicit)
__global__ void k_invert(float* __restrict__ p, int n) {
    int i = blockIdx.x * blockDim.x + threadIdx.x;
    if (i < n) p[i] = 1.0f / p[i];
}

// agg[i][f] = h[i][f] * deg_inv[i]   (self-loop term; also zero-inits agg)
__global__ void k_init_agg(const float* __restrict__ h, const float* __restrict__ dinv,
                           float* __restrict__ agg, long total, int F) {
    long idx = (long)blockIdx.x * blockDim.x + threadIdx.x;
    if (idx < total) agg[idx] = h[idx] * dinv[idx / F];
}

// agg[col[e]][f] += h[row[e]][f] * deg_inv[col[e]]   (one block per edge, F threads)
__global__ void k_edge_scatter(const float* __restrict__ h, const int* __restrict__ row,
                               const int* __restrict__ col, const float* __restrict__ dinv,
                               float* __restrict__ agg, int F) {
    int e = blockIdx.x;
    int f = threadIdx.x;
    int r = row[e];
    int c = col[e];
    float w = dinv[c];
    atomicAdd(&agg[(size_t)c * F + f], h[(size_t)r * F + f] * w);
}

// Zero-pad a [K, Fo] matrix to [K, FOP] columns.
__global__ void k_pad_w(const float* __restrict__ src, float* __restrict__ dst,
                        int K, int Fo, int FOP) {
    int idx = blockIdx.x * blockDim.x + threadIdx.x;
    if (idx < K * FOP) {
        int r = idx / FOP;
        int c = idx - r * FOP;
        dst[idx] = (c < Fo) ? src[r * Fo + c] : 0.0f;
    }
}

// Fused dense layer: out = act( A @ Wa + H @ Wh + bias )
// FO: compile-time padded output width (multiple of 16). Weights/bias are
// zero-padded to FO columns, so the K-loop needs no masking -> EXEC all-1s.
// Each wave computes a 64x16 block: 4 row-tiles reuse each B (weight) fragment.
// Row overrun (N not multiple of 64) handled by clamping A-row offsets to row
// N-1 (branchless) and guarding only the stores (after all WMMAs).
// A/H are indexed as v2f (8-byte) elements with 32-bit indices so the b64
// loads lower to global_load_b64 vDst, vOff, s[base] scale_offset.
// Loads for each k-step are staged into distinct locals BEFORE the WMMA burst
// so the compiler issues them as a clause and overlaps them with the WMMAs.
// act == 0: ReLU;  act == 1: sigmoid(ReLU(.)) (final layer of the reference).
template <int FO>
__global__ void __launch_bounds__(256) k_gcn_gemm(
    const float* __restrict__ A,    // agg  [N,K] row-major
    const float* __restrict__ H,    // h    [N,K] row-major
    const float* __restrict__ Wa,   // W_out  [K,FO] (padded)
    const float* __restrict__ Wh,   // W_root [K,FO] (padded)
    const float* __restrict__ bias, // [FO] (padded)
    float* __restrict__ out,        // [N,Fo_real]
    int N, int K, int Fo_real, int act)
{
    constexpr int COLT = FO / 16;
    const int lane = threadIdx.x & 31;
    const int wave = threadIdx.x >> 5;
    const int g = blockIdx.x * 8 + wave;          // (rowGroup, colTile) pairs
    const int rowGroups = (N + 63) >> 6;
    if (g >= rowGroups * COLT) return;            // wave-uniform exit

    const int rowGroup = g / COLT;
    const int colTile  = g - rowGroup * COLT;
    const int m0   = rowGroup << 6;               // first of 64 rows
    const int half = lane >> 4;                   // hi lanes: K+2 / M+8 halves
    const int ncol = (colTile << 4) + (lane & 15);
    const int khalf = half << 1;
    const int Nm1 = N - 1;

    // A/H viewed as v2f so b64 loads can use saddr + voffset scale_offset.
    const v2f* __restrict__ A2 = (const v2f*)A;
    const v2f* __restrict__ H2 = (const v2f*)H;
    const int K2 = K >> 1;                        // row stride in v2f units

    // 32-bit v2f-element offsets of each sub-tile's row (clamped to row N-1).
    unsigned offR2[4];
    #pragma unroll
    for (int r = 0; r < 4; ++r) {
        int mr = min(m0 + r * 16 + (lane & 15), Nm1);
        offR2[r] = (unsigned)mr * (unsigned)K2 + (unsigned)half;
    }
    const unsigned offW = (unsigned)khalf * FO + (unsigned)ncol;

    v8f acc[4] = {};
    #pragma unroll 2
    for (int k = 0; k < K; k += 4) {
        const unsigned wk = offW + (unsigned)k * FO;
        const unsigned k2 = (unsigned)(k >> 1);   // v2f-units advance

        // ---- load phase: 10 independent loads into distinct registers ----
        v2f bA, bH, a0[4], a1[4];
        bA.x = Wa[wk];  bA.y = Wa[wk + FO];
        bH.x = Wh[wk];  bH.y = Wh[wk + FO];
        #pragma unroll
        for (int r = 0; r < 4; ++r) {
            a0[r] = A2[offR2[r] + k2];
            a1[r] = H2[offR2[r] + k2];
        }

        // ---- compute phase: 8 WMMAs, waits can decrement load counter ----
        #pragma unroll
        for (int r = 0; r < 4; ++r) {
            acc[r] = __builtin_amdgcn_wmma_f32_16x16x4_f32(
                false, a0[r], false, bA, (short)0, acc[r], false, false);
            acc[r] = __builtin_amdgcn_wmma_f32_16x16x4_f32(
                false, a1[r], false, bH, (short)0, acc[r], false, false);
        }
    }

    const float b = bias[ncol];                   // padded: 0 beyond Fo_real
    const bool colOK = ncol < Fo_real;
    #pragma unroll
    for (int r = 0; r < 4; ++r) {
        const int rbase = m0 + r * 16 + half * 8; // C/D: VGPR i -> M=i (+8 hi)
        if (colOK && rbase < N) {                 // N multiple of 16 -> whole
            #pragma unroll                        // sub-tile valid iff rbase<N
            for (int i = 0; i < 8; ++i) {
                float v = acc[r][i] + b;
                v = fmaxf(v, 0.0f);                        // ReLU (all layers)
                if (act) v = 1.0f / (1.0f + expf(-v));     // final sigmoid
                out[(size_t)(rbase + i) * Fo_real + ncol] = v;
            }
        }
    }
}

extern "C" void kernel_launch(void* const* d_in, const int* in_sizes, int n_in,
                              void* d_out, int out_size, void* d_ws, size_t ws_size,
                              hipStream_t stream) {
    const float* x      = (const float*)d_in[0];
    const int*   ei     = (const int*)  d_in[1];
    const float* Wout0  = (const float*)d_in[2];
    const float* bout0  = (const float*)d_in[3];
    const float* Wroot0 = (const float*)d_in[4];
    const float* Wout1  = (const float*)d_in[5];
    const float* bout1  = (const float*)d_in[6];
    const float* Wroot1 = (const float*)d_in[7];
    const float* Wout2  = (const float*)d_in[8];
    const float* bout2  = (const float*)d_in[9];
    const float* Wroot2 = (const float*)d_in[10];

    const int N = in_sizes[0] / IN_CH;   // 50000
    const int E = in_sizes[1] / 2;       // 400000
    const int* row = ei;                 // edge_index[0] = source j
    const int* col = ei + E;             // edge_index[1] = target i

    // Workspace (floats):
    //   deg_inv[N] | agg[N*HID] | h0[N*HID] | h1[N*HID]
    //   | Wout2p[HID*OUT_PAD] | Wroot2p[HID*OUT_PAD] | b2p[OUT_PAD]
    float* deg    = (float*)d_ws;
    float* agg    = deg + N;
    float* h0     = agg + (size_t)N * HID;
    float* h1     = h0  + (size_t)N * HID;
    float* Wout2p  = h1 + (size_t)N * HID;
    float* Wroot2p = Wout2p  + (size_t)HID * OUT_PAD;
    float* b2p     = Wroot2p + (size_t)HID * OUT_PAD;

    const long NF = (long)N * HID;
    const int T = 256;
    const int nBlkN  = (N + T - 1) / T;
    const int nBlkE  = (E + T - 1) / T;
    const int nBlkNF = (int)((NF + T - 1) / T);

    // ---- degree (includes self loop) -> deg_inv ----
    k_fill1    <<<nBlkN, T, 0, stream>>>(deg, N);
    k_deg_accum<<<nBlkE, T, 0, stream>>>(col, deg, E);
    k_invert   <<<nBlkN, T, 0, stream>>>(deg, N);

    // ---- pad layer-2 weights/bias to 128 cols ----
    const int padElems = HID * OUT_PAD;
    k_pad_w<<<(padElems + T - 1) / T, T, 0, stream>>>(Wout2,  Wout2p,  HID, OUT_CH, OUT_PAD);
    k_pad_w<<<(padElems + T - 1) / T, T, 0, stream>>>(Wroot2, Wroot2p, HID, OUT_CH, OUT_PAD);
    k_pad_w<<<(OUT_PAD + T - 1) / T, T, 0, stream>>>(bout2,  b2p,     1,   OUT_CH, OUT_PAD);

    const int rowGroups = (N + 63) / 64;               // 782
    const int blk256 = (rowGroups * (HID / 16)     + 7) / 8;
    const int blk128 = (rowGroups * (OUT_PAD / 16) + 7) / 8;

    // ---- layer 0: x -> h0 ----
    k_init_agg    <<<nBlkNF, T, 0, stream>>>(x, deg, agg, NF, IN_CH);
    k_edge_scatter<<<E, IN_CH, 0, stream>>>(x, row, col, deg, agg, IN_CH);
    k_gcn_gemm<HID><<<blk256, 256, 0, stream>>>(agg, x, Wout0, Wroot0, bout0, h0,
                                                N, IN_CH, HID, /*act=*/0);

    // ---- layer 1: h0 -> h1 ----
    k_init_agg    <<<nBlkNF, T, 0, stream>>>(h0, deg, agg, NF, HID);
    k_edge_scatter<<<E, HID, 0, stream>>>(h0, row, col, deg, agg, HID);
    k_gcn_gemm<HID><<<blk256, 256, 0, stream>>>(agg, h0, Wout1, Wroot1, bout1, h1,
                                                N, HID, HID, /*act=*/0);

    // ---- layer 2: h1 -> sigmoid(relu(conv2)) -> d_out ----
    k_init_agg    <<<nBlkNF, T, 0, stream>>>(h1, deg, agg, NF, HID);
    k_edge_scatter<<<E, HID, 0, stream>>>(h1, row, col, deg, agg, HID);
    k_gcn_gemm<OUT_PAD><<<blk128, 256, 0, stream>>>(agg, h1, Wout2p, Wroot2p, b2p,
                                                    (float*)d_out,
                                                    N, HID, OUT_CH, /*act=*/1);
}